// GCN_27874337751415
// MI455X (gfx1250) — compile-verified
//
#include <hip/hip_runtime.h>
#include <hip/hip_bf16.h>

typedef __attribute__((ext_vector_type(2))) float v2f;
typedef __attribute__((ext_vector_type(8))) float v8f;

#define NNODES 50000
#define NEDGES 800000
#define NGRAPH 64

// CDNA5 async global->LDS path (ASYNCcnt-tracked), with sync fallback
#if defined(__has_builtin)
#if __has_builtin(__builtin_amdgcn_global_load_async_to_lds_b128)
#define GCN_HAVE_ASYNC_LDS 1
#endif
#endif

#ifdef GCN_HAVE_ASYNC_LDS
typedef __attribute__((ext_vector_type(4))) int gcn_v4i;
typedef __attribute__((address_space(1))) gcn_v4i* gcn_gvp;   // global (AS1) int4*
typedef __attribute__((address_space(3))) gcn_v4i* gcn_lvp;   // LDS (AS3) int4*
static __device__ __forceinline__ void gcn_wait_async_le(int n) {
#if __has_builtin(__builtin_amdgcn_s_wait_asynccnt)
  if (n == 0) __builtin_amdgcn_s_wait_asynccnt(0);
  else        __builtin_amdgcn_s_wait_asynccnt(4);
#else
  if (n == 0) asm volatile("s_wait_asynccnt 0x0" ::: "memory");
  else        asm volatile("s_wait_asynccnt 0x4" ::: "memory");
#endif
}
#endif

// ---------------------------------------------------------------- utilities
__global__ void gcn_fill_kernel(float* p, int n, float v) {
  int t = blockIdx.x * blockDim.x + threadIdx.x;
  if (t < n) p[t] = v;
}

__global__ void gcn_edge_deg_kernel(const int* __restrict__ ei, float* deg, int E) {
  int e = blockIdx.x * blockDim.x + threadIdx.x;
  if (e < E) atomicAdd(&deg[ei[E + e]], 1.0f);   // dst degrees; self-loop preloaded as 1.0
}

__global__ void gcn_rsqrt_kernel(float* d, int n) {
  int t = blockIdx.x * blockDim.x + threadIdx.x;
  if (t < n) d[t] = rsqrtf(d[t]);                // deg >= 1 always (self-loops)
}

// ------------------------------------------------- WMMA fp32 GEMM: C = A @ W
// A: [M,K] row-major (M multiple of 16), W: [K,Nout], C: [M,Nout]
// One wave = 16 rows x 64 cols (4 fp32 WMMA accumulators reusing one A frag).
// W panel double-buffered through LDS via async global->LDS B128 copies.
// Row stride 72 dwords: 2*72 mod 64 = 16 -> half-waves hit disjoint bank sets.
#define GCN_KC 64
#define GCN_SSTRIDE 72
#define GCN_CHUNK_FLOATS (GCN_KC * GCN_SSTRIDE)

__global__ void __launch_bounds__(256)
gcn_gemm_wmma_kernel(const float* __restrict__ A, const float* __restrict__ W,
                     float* __restrict__ C, int M, int K, int Nout) {
  __shared__ __align__(16) float sW[2 * GCN_CHUNK_FLOATS];
  const int colBase = blockIdx.y * 64;
  const int wave = threadIdx.x >> 5;
  const int lane = threadIdx.x & 31;
  const int mtile = blockIdx.x * 8 + wave;
  const bool active = (mtile * 16) < M;
  const int half = lane >> 4;                    // 0: lanes 0-15, 1: lanes 16-31
  const int koff = half * 2;                     // fp32 A/B frag: K = half*2 + {0,1}
  const int ncol = lane & 15;
  const int mrow = (active ? mtile : 0) * 16 + ncol;

  v8f acc0 = {}, acc1 = {}, acc2 = {}, acc3 = {};
  const float* arow = A + (size_t)mrow * K;
  const int nchunks = K / GCN_KC;

#ifdef GCN_HAVE_ASYNC_LDS
  // each thread copies 4 B128 chunks per 64x64 W panel (16 b128 per K-row)
  const int crow = threadIdx.x >> 4;             // base row
  const int ccol = (threadIdx.x & 15) * 4;       // float column
  auto issue_chunk = [&](int c, int buf) {
    const float* gsrc = W + (size_t)(c * GCN_KC) * Nout + colBase + ccol;
    float* ldst = &sW[buf * GCN_CHUNK_FLOATS + crow * GCN_SSTRIDE + ccol];
    for (int i = 0; i < 4; ++i) {               // rows crow, crow+16, +32, +48
      __builtin_amdgcn_global_load_async_to_lds_b128(
          (gcn_gvp)(gsrc + (size_t)(i * 16) * Nout),
          (gcn_lvp)(ldst + i * 16 * GCN_SSTRIDE), 0, 0);
    }
  };
  issue_chunk(0, 0);
  for (int c = 0; c < nchunks; ++c) {
    const int cur = c & 1;
    const bool more = (c + 1) < nchunks;
    if (more) issue_chunk(c + 1, cur ^ 1);
    gcn_wait_async_le(more ? 4 : 0);             // chunk c's copies complete (in-order)
    __syncthreads();
    if (active) {
      const float* sb = &sW[cur * GCN_CHUNK_FLOATS];
      const float* ar = arow + c * GCN_KC;
      for (int kb = 0; kb < GCN_KC; kb += 4) {
        const float2 av = *reinterpret_cast<const float2*>(ar + kb + koff);
        v2f a; a.x = av.x; a.y = av.y;
        const float* w0 = &sb[(kb + koff) * GCN_SSTRIDE];
        const float* w1 = &sb[(kb + koff + 1) * GCN_SSTRIDE];
        v2f b;
        b.x = w0[ncol];      b.y = w1[ncol];
        acc0 = __builtin_amdgcn_wmma_f32_16x16x4_f32(false, a, false, b, (short)0, acc0, false, false);
        b.x = w0[16 + ncol]; b.y = w1[16 + ncol];
        acc1 = __builtin_amdgcn_wmma_f32_16x16x4_f32(false, a, false, b, (short)0, acc1, false, false);
        b.x = w0[32 + ncol]; b.y = w1[32 + ncol];
        acc2 = __builtin_amdgcn_wmma_f32_16x16x4_f32(false, a, false, b, (short)0, acc2, false, false);
        b.x = w0[48 + ncol]; b.y = w1[48 + ncol];
        acc3 = __builtin_amdgcn_wmma_f32_16x16x4_f32(false, a, false, b, (short)0, acc3, false, false);
      }
    }
    __syncthreads();                             // reads done before buf refilled at c+2
  }
#else
  for (int c = 0; c < nchunks; ++c) {
    for (int idx = threadIdx.x; idx < GCN_KC * 64; idx += 256) {
      int kk = idx >> 6, nn = idx & 63;
      sW[kk * GCN_SSTRIDE + nn] = W[(size_t)(c * GCN_KC + kk) * Nout + colBase + nn];
    }
    __syncthreads();
    if (active) {
      const float* ar = arow + c * GCN_KC;
      for (int kb = 0; kb < GCN_KC; kb += 4) {
        const float2 av = *reinterpret_cast<const float2*>(ar + kb + koff);
        v2f a; a.x = av.x; a.y = av.y;
        const float* w0 = &sW[(kb + koff) * GCN_SSTRIDE];
        const float* w1 = &sW[(kb + koff + 1) * GCN_SSTRIDE];
        v2f b;
        b.x = w0[ncol];      b.y = w1[ncol];
        acc0 = __builtin_amdgcn_wmma_f32_16x16x4_f32(false, a, false, b, (short)0, acc0, false, false);
        b.x = w0[16 + ncol]; b.y = w1[16 + ncol];
        acc1 = __builtin_amdgcn_wmma_f32_16x16x4_f32(false, a, false, b, (short)0, acc1, false, false);
        b.x = w0[32 + ncol]; b.y = w1[32 + ncol];
        acc2 = __builtin_amdgcn_wmma_f32_16x16x4_f32(false, a, false, b, (short)0, acc2, false, false);
        b.x = w0[48 + ncol]; b.y = w1[48 + ncol];
        acc3 = __builtin_amdgcn_wmma_f32_16x16x4_f32(false, a, false, b, (short)0, acc3, false, false);
      }
    }
    __syncthreads();
  }
#endif

  if (active) {                                   // C/D layout: M = half*8 + r, N = ncol
    const int mstore = mtile * 16 + half * 8;
    for (int r = 0; r < 8; ++r) {
      float* crow2 = C + (size_t)(mstore + r) * Nout + colBase;
      crow2[ncol]      = acc0[r];
      crow2[16 + ncol] = acc1[r];
      crow2[32 + ncol] = acc2[r];
      crow2[48 + ncol] = acc3[r];
    }
  }
}

// ----------------------------------------------- aggregation (A_hat message)
// out = h * dinv^2 (self-loop) + bias ; then edge scatter adds h[src]*norm
__global__ void gcn_agg_init_kernel(const float* __restrict__ h, const float* __restrict__ dinv,
                                    const float* __restrict__ bias, float* out, int n, int fshift) {
  int t = blockIdx.x * blockDim.x + threadIdx.x;
  if (t < n) {
    int i = t >> fshift;
    int f = t & ((1 << fshift) - 1);
    float di = dinv[i];
    out[t] = h[t] * di * di + bias[f];
  }
}

__global__ void gcn_agg_edge_kernel(const int* __restrict__ ei, const float* __restrict__ h,
                                    const float* __restrict__ dinv, float* out, int E, int fshift) {
  int t = blockIdx.x * blockDim.x + threadIdx.x;
  const int cshift = fshift - 2;                  // float4 chunks per edge
  const int total = E << cshift;
  if (t >= total) return;
  int e = t >> cshift;
  int c = (t & ((1 << cshift) - 1)) << 2;
  int s = ei[e];
  int d = ei[E + e];
  float nrm = dinv[s] * dinv[d];
  const float4 v = *reinterpret_cast<const float4*>(h + ((size_t)s << fshift) + c);
  float* o = out + ((size_t)d << fshift) + c;
  atomicAdd(o + 0, v.x * nrm);
  atomicAdd(o + 1, v.y * nrm);
  atomicAdd(o + 2, v.z * nrm);
  atomicAdd(o + 3, v.w * nrm);
}

__global__ void gcn_relu_kernel(float* p, int n) {
  int t = blockIdx.x * blockDim.x + threadIdx.x;
  if (t < n) p[t] = fmaxf(p[t], 0.0f);
}

// ------------------------------------------------------------ pool and head
__global__ void gcn_pool_kernel(const float* __restrict__ h, const int* __restrict__ batch,
                                float* psum, float* pcnt, int n) {
  int t = blockIdx.x * blockDim.x + threadIdx.x;
  if (t < n) {
    int i = t >> 6, f = t & 63;
    int b = batch[i];
    atomicAdd(&psum[b * 64 + f], h[t]);
    if (f == 0) atomicAdd(&pcnt[b], 1.0f);
  }
}

__global__ void gcn_head_kernel(const float* __restrict__ psum, const float* __restrict__ pcnt,
                                const float* __restrict__ Wl, const float* __restrict__ bl,
                                float* out) {
  int t = threadIdx.x;                           // 128 threads = 64 graphs x 2 outputs
  int g = t >> 1, j = t & 1;
  float c = fmaxf(pcnt[g], 1.0f);
  float acc = bl[j];
  for (int f = 0; f < 64; ++f)
    acc += (psum[g * 64 + f] / c) * Wl[f * 2 + j];
  out[g * 2 + j] = acc;
}

// ----------------------------------------------------------------- launcher
extern "C" void kernel_launch(void* const* d_in, const int* in_sizes, int n_in,
                              void* d_out, int out_size, void* d_ws, size_t ws_size,
                              hipStream_t stream) {
  (void)in_sizes; (void)n_in; (void)out_size; (void)ws_size;
  const float* x   = (const float*)d_in[0];
  const int*   ei  = (const int*)d_in[1];
  const int*   bat = (const int*)d_in[2];
  const float* W1  = (const float*)d_in[3];
  const float* b1  = (const float*)d_in[4];
  const float* W2  = (const float*)d_in[5];
  const float* b2  = (const float*)d_in[6];
  const float* W3  = (const float*)d_in[7];
  const float* b3  = (const float*)d_in[8];
  const float* Wl  = (const float*)d_in[9];
  const float* bl  = (const float*)d_in[10];
  float* out = (float*)d_out;

  const int N = NNODES, E = NEDGES, G = NGRAPH, T = 256;
  float* dinv = (float*)d_ws;                     // N
  float* bufH = dinv + N;                         // N*256 (GEMM out)
  float* bufA = bufH + (size_t)N * 256;           // N*256 (aggregated / next input)
  float* psum = bufA + (size_t)N * 256;           // G*64
  float* pcnt = psum + G * 64;                    // G

  auto nb = [](int n, int t) { return (n + t - 1) / t; };
  const int mblocks = nb(N / 16, 8);              // 3125 m-tiles -> 391 blocks of 8 waves

  // normalized degrees
  gcn_fill_kernel<<<nb(N, T), T, 0, stream>>>(dinv, N, 1.0f);
  gcn_edge_deg_kernel<<<nb(E, T), T, 0, stream>>>(ei, dinv, E);
  gcn_rsqrt_kernel<<<nb(N, T), T, 0, stream>>>(dinv, N);

  // layer 1: 512 -> 256, ReLU
  gcn_gemm_wmma_kernel<<<dim3(mblocks, 4), T, 0, stream>>>(x, W1, bufH, N, 512, 256);
  gcn_agg_init_kernel<<<nb(N * 256, T), T, 0, stream>>>(bufH, dinv, b1, bufA, N * 256, 8);
  gcn_agg_edge_kernel<<<nb(E << 6, T), T, 0, stream>>>(ei, bufH, dinv, bufA, E, 8);
  gcn_relu_kernel<<<nb(N * 256, T), T, 0, stream>>>(bufA, N * 256);

  // layer 2: 256 -> 128, ReLU
  gcn_gemm_wmma_kernel<<<dim3(mblocks, 2), T, 0, stream>>>(bufA, W2, bufH, N, 256, 128);
  gcn_agg_init_kernel<<<nb(N * 128, T), T, 0, stream>>>(bufH, dinv, b2, bufA, N * 128, 7);
  gcn_agg_edge_kernel<<<nb(E << 5, T), T, 0, stream>>>(ei, bufH, dinv, bufA, E, 7);
  gcn_relu_kernel<<<nb(N * 128, T), T, 0, stream>>>(bufA, N * 128);

  // layer 3: 128 -> 64 (no ReLU)
  gcn_gemm_wmma_kernel<<<dim3(mblocks, 1), T, 0, stream>>>(bufA, W3, bufH, N, 128, 64);
  gcn_agg_init_kernel<<<nb(N * 64, T), T, 0, stream>>>(bufH, dinv, b3, bufA, N * 64, 6);
  gcn_agg_edge_kernel<<<nb(E << 4, T), T, 0, stream>>>(ei, bufH, dinv, bufA, E, 6);

  // mean pool + linear head
  gcn_fill_kernel<<<nb(G * 64 + G, T), T, 0, stream>>>(psum, G * 64 + G, 0.0f);
  gcn_pool_kernel<<<nb(N * 64, T), T, 0, stream>>>(bufA, bat, psum, pcnt, N * 64);
  gcn_head_kernel<<<1, 128, 0, stream>>>(psum, pcnt, Wl, bl, out);
}